// Augmenter_6416681140826
// MI455X (gfx1250) — compile-verified
//
#include <hip/hip_runtime.h>
#include <hip/hip_bf16.h>

#define CROPSZ 512
#define GRIDN  8
#define NBINS  256
#define TILE   64           // CROPSZ / GRIDN
#define TP     (TILE*TILE)  // 4096 pixels per tile
#define H_IN   640
#define W_IN   640
#define NB     64           // batch

typedef __attribute__((ext_vector_type(2))) float v2f;
typedef __attribute__((ext_vector_type(8))) float v8f;

// Source gather with hflip/vflip applied before crop (matches reference:
// flip whole image, then dynamic_slice at (offy, offx)).
__device__ __forceinline__ float load_src(const float* __restrict__ x, int b,
                                          int i, int j, int hf, int vf,
                                          int oy, int ox) {
    int sy = oy + i;
    int sx = ox + j;
    if (vf) sy = H_IN - 1 - sy;
    if (hf) sx = W_IN - 1 - sx;
    return x[(size_t)b * (H_IN * W_IN) + (size_t)sy * W_IN + sx];
}

// ---------------------------------------------------------------------------
// Kernel 1: per-tile histogram -> clip/redistribute -> cumsum LUT via WMMA.
// One block per (image, tile): 64 * 64 = 4096 blocks, 256 threads each.
// The 256-bin inclusive scan is computed as LUT = Lstrict*T + H*U using
// eight chained V_WMMA_F32_16X16X4_F32 ops in wave 0.
// ---------------------------------------------------------------------------
__global__ void __launch_bounds__(256) clahe_lut_kernel(
    const float* __restrict__ x, const int* __restrict__ hflip,
    const int* __restrict__ vflip, const int* __restrict__ offy,
    const int* __restrict__ offx, const int* __restrict__ apply,
    float* __restrict__ lutG) {
    const int b    = blockIdx.x >> 6;
    const int tile = blockIdx.x & 63;
    if (!apply[b]) return;  // uniform per block: LUT never read for these images

    const int tyi = tile >> 3;
    const int txi = tile & 7;
    const int hf = hflip[b], vf = vflip[b], oy = offy[b], ox = offx[b];

    __shared__ unsigned int histU[NBINS];
    __shared__ float hist2[NBINS];
    __shared__ float sred[NBINS];
    __shared__ float ldsT[16];

    const int tid = threadIdx.x;
    histU[tid] = 0u;
    __syncthreads();

    // Histogram: 16 pixels per thread, coalesced 64-float rows.
    for (int p = tid; p < TP; p += 256) {
        const int i = p >> 6;
        const int j = p & 63;
        float v = load_src(x, b, tyi * TILE + i, txi * TILE + j, hf, vf, oy, ox);
        int bin = (int)(v * 256.0f);
        bin = bin < 0 ? 0 : (bin > 255 ? 255 : bin);
        atomicAdd(&histU[bin], 1u);  // ds_add_u32: exact + deterministic
    }
    __syncthreads();

    // Clip + deterministic excess reduction (fixed-order tree).
    const float clipv = 12.8f;  // max(0.8*4096/256, 1)
    const float h = (float)histU[tid];
    const float c = fminf(h, clipv);
    sred[tid] = h - c;
    __syncthreads();
    for (int s = 128; s > 0; s >>= 1) {
        if (tid < s) sred[tid] += sred[tid + s];
        __syncthreads();
    }
    hist2[tid] = c + sred[0] * (1.0f / NBINS);
    __syncthreads();

    // Wave 0: 256-bin inclusive scan as two 16x16 matmuls on the WMMA unit.
    const int  lane = tid & 31;
    const bool w0   = tid < 32;          // full wave -> EXEC all ones for WMMA
    const int  m    = (lane < 16) ? lane : (lane - 16);
    const int  koff = (lane < 16) ? 0 : 2;  // A/B K-slot split across lane halves
    v8f pm = {};

    if (w0) {
        // P = H x U   (H[r][k] = hist2[16r+k], U upper-triangular ones)
        for (int kb = 0; kb < 4; ++kb) {
            v2f a, bb;
            a.x = hist2[16 * m + 4 * kb + koff];
            a.y = hist2[16 * m + 4 * kb + koff + 1];
            bb.x = ((4 * kb + koff)     <= m) ? 1.0f : 0.0f;
            bb.y = ((4 * kb + koff + 1) <= m) ? 1.0f : 0.0f;
            pm = __builtin_amdgcn_wmma_f32_16x16x4_f32(false, a, false, bb,
                                                       (short)0, pm, false, false);
        }
        // Row totals T[r] = P[r][15] live in lanes 15 (rows 0-7) / 31 (rows 8-15).
        if (lane == 15) {
            for (int i = 0; i < 8; ++i) ldsT[i] = pm[i];
        }
        if (lane == 31) {
            for (int i = 0; i < 8; ++i) ldsT[8 + i] = pm[i];
        }
    }
    __syncthreads();  // block-uniform barrier publishes ldsT

    if (w0) {
        // LUT = Lstrict x broadcast(T) + P   (accumulate into pm)
        for (int kb = 0; kb < 4; ++kb) {
            v2f a, bb;
            a.x = ((4 * kb + koff)     < m) ? 1.0f : 0.0f;
            a.y = ((4 * kb + koff + 1) < m) ? 1.0f : 0.0f;
            bb.x = ldsT[4 * kb + koff];
            bb.y = ldsT[4 * kb + koff + 1];
            pm = __builtin_amdgcn_wmma_f32_16x16x4_f32(false, a, false, bb,
                                                       (short)0, pm, false, false);
        }
        const float scale = (NBINS - 1.0f) / (float)TP;  // 255/4096
        float* lut = lutG + ((size_t)b * 64 + tile) * NBINS;
        for (int i = 0; i < 8; ++i) {
            const int row = (lane < 16) ? i : (i + 8);  // C layout: vgpr i -> M=i / i+8
            float vv = pm[i] * scale;
            vv = fminf(fmaxf(vv, 0.0f), 255.0f);
            lut[16 * row + m] = vv;
        }
    }
}

// ---------------------------------------------------------------------------
// Kernel 2: per-pixel gather + bilinear LUT interpolation. LUT table (4 MB)
// stays L2-resident; traffic is ~67 MB read + 67 MB write -> HBM-bound.
// ---------------------------------------------------------------------------
__global__ void __launch_bounds__(256) clahe_apply_kernel(
    const float* __restrict__ x, const int* __restrict__ hflip,
    const int* __restrict__ vflip, const int* __restrict__ offy,
    const int* __restrict__ offx, const int* __restrict__ apply,
    const float* __restrict__ lutG, float* __restrict__ out) {
    const size_t idx = (size_t)blockIdx.x * blockDim.x + threadIdx.x;
    const size_t total = (size_t)NB * CROPSZ * CROPSZ;
    if (idx >= total) return;

    const int j = (int)(idx & (CROPSZ - 1));
    const int i = (int)((idx >> 9) & (CROPSZ - 1));
    const int b = (int)(idx >> 18);

    const float v = load_src(x, b, i, j, hflip[b], vflip[b], offy[b], offx[b]);
    float res;
    if (!apply[b]) {
        res = v;
    } else {
        int bin = (int)(v * 256.0f);
        bin = bin < 0 ? 0 : (bin > 255 ? 255 : bin);

        const float tyc = (i + 0.5f) * (1.0f / TILE) - 0.5f;
        const float txc = (j + 0.5f) * (1.0f / TILE) - 0.5f;
        const float y0f = floorf(tyc), x0f = floorf(txc);
        const float fy = tyc - y0f, fx = txc - x0f;
        const int y0 = (int)fminf(fmaxf(y0f, 0.0f), 7.0f);
        const int y1 = (int)fminf(fmaxf(y0f + 1.0f, 0.0f), 7.0f);
        const int x0 = (int)fminf(fmaxf(x0f, 0.0f), 7.0f);
        const int x1 = (int)fminf(fmaxf(x0f + 1.0f, 0.0f), 7.0f);

        const float* lutB = lutG + (size_t)b * 64 * NBINS;
        const float v00 = lutB[(y0 * 8 + x0) * NBINS + bin];
        const float v01 = lutB[(y0 * 8 + x1) * NBINS + bin];
        const float v10 = lutB[(y1 * 8 + x0) * NBINS + bin];
        const float v11 = lutB[(y1 * 8 + x1) * NBINS + bin];

        const float top = (1.0f - fx) * v00 + fx * v01;
        const float bot = (1.0f - fx) * v10 + fx * v11;
        res = ((1.0f - fy) * top + fy * bot) * (1.0f / (NBINS - 1.0f));
    }
    out[idx] = res;
}

extern "C" void kernel_launch(void* const* d_in, const int* in_sizes, int n_in,
                              void* d_out, int out_size, void* d_ws, size_t ws_size,
                              hipStream_t stream) {
    const float* x     = (const float*)d_in[0];
    const int* hflip   = (const int*)d_in[1];
    const int* vflip   = (const int*)d_in[2];
    const int* offy    = (const int*)d_in[3];
    const int* offx    = (const int*)d_in[4];
    const int* apply   = (const int*)d_in[5];
    float* out  = (float*)d_out;
    float* lutG = (float*)d_ws;  // 64 images * 64 tiles * 256 bins * 4B = 4 MB

    // Pass 1: per-tile histogram + WMMA prefix-scan LUT build.
    clahe_lut_kernel<<<NB * GRIDN * GRIDN, 256, 0, stream>>>(
        x, hflip, vflip, offy, offx, apply, lutG);

    // Pass 2: per-pixel gather + bilinear LUT blend.
    const size_t total = (size_t)NB * CROPSZ * CROPSZ;
    const int blocks = (int)((total + 255) / 256);
    clahe_apply_kernel<<<blocks, 256, 0, stream>>>(
        x, hflip, vflip, offy, offx, apply, lutG, out);
}